// LegislativeGraphEncoder_61607010893927
// MI455X (gfx1250) — compile-verified
//
#include <hip/hip_runtime.h>
#include <math.h>

typedef __attribute__((ext_vector_type(2))) float v2f;
typedef __attribute__((ext_vector_type(8))) float v8f;

#define NLEG   20000
#define NBILL  50000
#define NBV    100000
#define NDON   50000
#define NLOB   20000
#define EDON   300000
#define ELOB   150000
#define EVER   150000
#define EVOTE  200000
#define DD     192
#define DHH    96

// ---------------------------------------------------------------- helpers
__device__ __forceinline__ float gelu_exact(float x) {
    return 0.5f * x * (1.0f + erff(x * 0.70710678118654752f));
}
__device__ __forceinline__ unsigned flt_flip(float f) {
    unsigned u = __float_as_uint(f);
    return (u >> 31) ? ~u : (u | 0x80000000u);
}
__device__ __forceinline__ float flt_unflip(unsigned s) {
    return __uint_as_float((s >> 31) ? (s & 0x7FFFFFFFu) : ~s);
}
__device__ __forceinline__ v8f wmma4(v2f a, v2f b, v8f c) {
    // V_WMMA_F32_16X16X4_F32 : exact fp32, D = A(16x4) * B(4x16) + C(16x16)
    return __builtin_amdgcn_wmma_f32_16x16x4_f32(false, a, false, b, (short)0, c,
                                                 false, false);
}

// ---------------------------------------------------------------- elementwise
__global__ void fill_f32(float* p, float v, int n) {
    int i = blockIdx.x * 256 + threadIdx.x;
    if (i < n) p[i] = v;
}
__global__ void fill_u32(unsigned* p, unsigned v, int n) {
    int i = blockIdx.x * 256 + threadIdx.x;
    if (i < n) p[i] = v;
}
__global__ void add2_f32(float* __restrict__ d, const float* __restrict__ a,
                         const float* __restrict__ b, int n) {
    int i = blockIdx.x * 256 + threadIdx.x;
    if (i < n) d[i] = a[i] + b[i];
}
__global__ void gelu_f32(float* __restrict__ d, const float* __restrict__ a, int n) {
    int i = blockIdx.x * 256 + threadIdx.x;
    if (i < n) d[i] = gelu_exact(a[i]);
}

// ---------------------------------------------------------------- generic WMMA GEMM
// One wave computes one 16x16 tile of Y = X @ W (+ bias), with optional
// gated-skip / residual / accumulate epilogue.  All dims mult of 16, K of 4.
// Hot loop is branch-free: b64 A load + 2x b32 B load + v_wmma per K-step.
__global__ void gemm16_wmma(const float* __restrict__ X, const float* __restrict__ W,
                            int ldw, const float* __restrict__ bias,
                            float* __restrict__ Y, int ldy,
                            const float* __restrict__ gate,
                            const float* __restrict__ skipx,
                            const float* __restrict__ extra, int accum, int K,
                            int ldx) {
    const int lane = threadIdx.x & 31;
    const int half = lane >> 4;   // 0: K rows {0,1}, 1: K rows {2,3}
    const int l15  = lane & 15;
    const int kg   = half * 2;
    const int m0   = blockIdx.x * 16;
    const int n0   = blockIdx.y * 16;

    const float* xr = X + (size_t)(m0 + l15) * ldx + kg;
    const float* wr = W + (size_t)kg * ldw + n0 + l15;

    v8f acc = {0.f, 0.f, 0.f, 0.f, 0.f, 0.f, 0.f, 0.f};
#pragma unroll 4
    for (int k = 0; k < K; k += 4) {
        v2f a = *(const v2f*)(xr + k);
        v2f b;
        b.x = wr[(size_t)k * ldw];
        b.y = wr[(size_t)(k + 1) * ldw];
        acc = wmma4(a, b, acc);
    }

    float ga = 1.0f;
    if (gate) ga = 1.0f / (1.0f + expf(-gate[0]));
    const int rbase = half * 8;
    for (int r = 0; r < 8; ++r) {
        int    m  = m0 + rbase + r;
        int    n  = n0 + l15;
        size_t yi = (size_t)m * ldy + n;
        float  v  = acc[r];
        if (bias) v += bias[n];
        if (gate) v = ga * v + (1.0f - ga) * skipx[yi];
        if (extra) v += extra[yi];
        if (accum) Y[yi] += v; else Y[yi] = v;
    }
}

// ---------------------------------------------------------------- temporal MLP scatter
// one block (192 thr) per edge: et = relu(ts*W1+b1) @ W2 + b2 ; atomic scatter-sum
__global__ void temporal_edge(const float* __restrict__ ts, const int* __restrict__ src,
                              int E, const float* __restrict__ W1,
                              const float* __restrict__ b1, const float* __restrict__ W2,
                              const float* __restrict__ b2, float* __restrict__ ta,
                              float* __restrict__ cnt) {
    __shared__ float h[48];
    int e = blockIdx.x;
    if (e >= E) return;
    float t = ts[e];
    if (threadIdx.x < 48) h[threadIdx.x] = fmaxf(t * W1[threadIdx.x] + b1[threadIdx.x], 0.f);
    __syncthreads();
    float acc = b2[threadIdx.x];
#pragma unroll 8
    for (int j = 0; j < 48; ++j) acc += h[j] * W2[j * DD + threadIdx.x];
    int s = src[e];
    atomicAdd(&ta[(size_t)s * DD + threadIdx.x], acc);
    if (threadIdx.x == 0) atomicAdd(&cnt[s], 1.0f);
}
__global__ void ta_finalize(float* __restrict__ ta, const float* __restrict__ cnt, int n) {
    int i = blockIdx.x * 256 + threadIdx.x;
    if (i >= n * DD) return;
    float v = ta[i] / fmaxf(cnt[i / DD], 1.0f);
    if (!isfinite(v)) v = 0.0f;
    ta[i] = v;
}

// ---------------------------------------------------------------- edge attention
// one wave per edge: alpha[e,h] = p_rel[h]/sqrt(96) * <K[src,h,:], Q[dst,h,:]>
__global__ void edge_alpha(const int* __restrict__ ei, int E, const float* __restrict__ Kb,
                           const float* __restrict__ Qb, const float* __restrict__ prel,
                           float* __restrict__ alpha, unsigned* __restrict__ mx) {
    int widx = blockIdx.x * (blockDim.x >> 5) + (threadIdx.x >> 5);
    int lane = threadIdx.x & 31;
    if (widx >= E) return;
    int s = ei[widx], d = ei[E + widx];
    const float* kr = Kb + (size_t)s * DD;
    const float* qr = Qb + (size_t)d * DD;
    for (int h = 0; h < 2; ++h) {
        int   o = h * DHH + lane;
        float p = kr[o] * qr[o] + kr[o + 32] * qr[o + 32] + kr[o + 64] * qr[o + 64];
        for (int off = 16; off > 0; off >>= 1) p += __shfl_down(p, off, 32);
        if (lane == 0) {
            float a = p * prel[h] * 0.1020620726159658f;  // 1/sqrt(96)
            alpha[(size_t)widx * 2 + h] = a;
            atomicMax(&mx[(size_t)d * 2 + h], flt_flip(a));
        }
    }
}
__global__ void edge_expsum(const int* __restrict__ ei, int E, float* __restrict__ alpha,
                            const unsigned* __restrict__ mx, float* __restrict__ den) {
    int i = blockIdx.x * 256 + threadIdx.x;
    if (i >= E * 2) return;
    int   e = i >> 1, h = i & 1;
    int   d = ei[E + e];
    float m = flt_unflip(mx[(size_t)d * 2 + h]);
    if (!isfinite(m)) m = 0.0f;
    float ex = expf(alpha[i] - m);
    alpha[i] = ex;
    atomicAdd(&den[(size_t)d * 2 + h], ex);
}
__global__ void edge_agg(const int* __restrict__ ei, int E, const float* __restrict__ alpha,
                         const float* __restrict__ den, const float* __restrict__ Vb,
                         float* __restrict__ agg) {
    int i = blockIdx.x * 256 + threadIdx.x;
    if (i >= E * DD) return;
    int   e = i / DD, dd = i - e * DD;
    int   h = dd >= DHH ? 1 : 0;
    int   s = ei[e], d = ei[E + e];
    float w = alpha[(size_t)e * 2 + h] / fmaxf(den[(size_t)d * 2 + h], 1e-16f);
    atomicAdd(&agg[(size_t)d * DD + dd], Vb[(size_t)s * DD + dd] * w);
}

// ---------------------------------------------------------------- polarity conv (WMMA)
// 16 edges / block, 4 waves. GEMM1 K=384 -> relu LDS -> GEMM2 K=192 -> scaled scatter.
__global__ void polarity_conv(const float* __restrict__ attr, const int* __restrict__ ei,
                              int E, const float* __restrict__ W1,
                              const float* __restrict__ b1, const float* __restrict__ W2,
                              const float* __restrict__ b2, float* __restrict__ sum,
                              float* __restrict__ cnt) {
    __shared__ float hsh[16][193];
    __shared__ float pol[16];
    __shared__ int   dsts[16];
    const int e0   = blockIdx.x * 16;
    const int wave = threadIdx.x >> 5;
    const int lane = threadIdx.x & 31;
    const int half = lane >> 4;
    const int l15  = lane & 15;
    const int kg   = half * 2;

    if (threadIdx.x < 16) {
        int e             = e0 + threadIdx.x;
        pol[threadIdx.x]  = fminf(fmaxf(attr[(size_t)e * 385], 0.0f), 1.0f) + 0.01f;
        dsts[threadIdx.x] = ei[E + e];
    }
    __syncthreads();

    const float* arow = attr + (size_t)(e0 + l15) * 385 + 1 + kg;  // 384 features
    // GEMM1: (16 x 384) @ (384 x 192) -> relu -> LDS
    for (int i = 0; i < 3; ++i) {
        int n0  = (wave * 3 + i) * 16;
        const float* wr = W1 + (size_t)kg * DD + n0 + l15;
        v8f acc = {0.f, 0.f, 0.f, 0.f, 0.f, 0.f, 0.f, 0.f};
#pragma unroll 4
        for (int k = 0; k < 384; k += 4) {
            v2f a = *(const v2f*)(arow + k);
            v2f b;
            b.x = wr[(size_t)k * DD];
            b.y = wr[(size_t)(k + 1) * DD];
            acc = wmma4(a, b, acc);
        }
        int rbase = half * 8;
        for (int r = 0; r < 8; ++r)
            hsh[rbase + r][n0 + l15] = fmaxf(acc[r] + b1[n0 + l15], 0.0f);
    }
    __syncthreads();
    // GEMM2: (16 x 192) @ (192 x 192) -> scale by pol -> atomic scatter to sum[dst]
    for (int i = 0; i < 3; ++i) {
        int n0  = (wave * 3 + i) * 16;
        const float* wr = W2 + (size_t)kg * DD + n0 + l15;
        v8f acc = {0.f, 0.f, 0.f, 0.f, 0.f, 0.f, 0.f, 0.f};
#pragma unroll 4
        for (int k = 0; k < DD; k += 4) {
            v2f a; a.x = hsh[l15][k + kg]; a.y = hsh[l15][k + kg + 1];
            v2f b;
            b.x = wr[(size_t)k * DD];
            b.y = wr[(size_t)(k + 1) * DD];
            acc = wmma4(a, b, acc);
        }
        int rbase = half * 8;
        for (int r = 0; r < 8; ++r) {
            int   m = rbase + r;
            int   n = n0 + l15;
            float v = (acc[r] + b2[n]) * pol[m];
            atomicAdd(&sum[(size_t)dsts[m] * DD + n], v);
        }
    }
    if (threadIdx.x < 16) atomicAdd(&cnt[dsts[threadIdx.x]], 1.0f);
}
__global__ void polarity_fin(float* __restrict__ out, const float* __restrict__ sum,
                             const float* __restrict__ cnt, int n) {
    int i = blockIdx.x * 256 + threadIdx.x;
    if (i >= n * DD) return;
    out[i] += sum[i] / fmaxf(cnt[i / DD], 1.0f);
}

// ---------------------------------------------------------------- host side
struct ConvP {
    const float *W_src, *b_src, *W_dst, *b_dst, *a_rel, *m_rel, *p_rel, *W_out, *b_out, *skip;
};

static void launch_gemm(hipStream_t s, const float* X, const float* W, int ldw,
                        const float* bias, float* Y, int ldy, const float* gate,
                        const float* skipx, const float* extra, int accum, int M, int N,
                        int K, int ldx) {
    dim3 g(M / 16, N / 16);
    gemm16_wmma<<<g, 32, 0, s>>>(X, W, ldw, bias, Y, ldy, gate, skipx, extra, accum, K,
                                 ldx);
}

static void run_hgt(hipStream_t s, const ConvP& p, const float* xsrc, const float* ta,
                    int Ns, const float* xdst, int Nd, const int* ei, int E,
                    float* Yregion, const float* extra, int accum, float* Kb, float* Vb,
                    float* Qb, float* AGG, unsigned* MAXA, float* DEN, float* ALPHA) {
    // src features (+ temporal adjustment) staged once in Qb (free until q-GEMM)
    const float* XS = xsrc;
    if (ta) {
        add2_f32<<<(Ns * DD + 255) / 256, 256, 0, s>>>(Qb, xsrc, ta, Ns * DD);
        XS = Qb;
    }
    // k_lin -> AGG(tmp); per-head * a_rel -> Kb
    launch_gemm(s, XS, p.W_src, 3 * DD, p.b_src, AGG, DD, 0, 0, 0, 0, Ns, DD, DD, DD);
    for (int h = 0; h < 2; ++h)
        launch_gemm(s, AGG + h * DHH, p.a_rel + h * DHH * DHH, DHH, 0, Kb + h * DHH, DD,
                    0, 0, 0, 0, Ns, DHH, DHH, DD);
    // v_lin -> AGG(tmp); per-head * m_rel -> Vb
    launch_gemm(s, XS, p.W_src + 2 * DD, 3 * DD, p.b_src + 2 * DD, AGG, DD, 0, 0, 0, 0,
                Ns, DD, DD, DD);
    for (int h = 0; h < 2; ++h)
        launch_gemm(s, AGG + h * DHH, p.m_rel + h * DHH * DHH, DHH, 0, Vb + h * DHH, DD,
                    0, 0, 0, 0, Ns, DHH, DHH, DD);
    // q = x_dst @ W_dst[:, D:2D] + b_dst[D:2D]   (overwrites XS staging)
    launch_gemm(s, xdst, p.W_dst + DD, 3 * DD, p.b_dst + DD, Qb, DD, 0, 0, 0, 0, Nd, DD,
                DD, DD);
    // attention (segment softmax over dst)
    fill_u32<<<(Nd * 2 + 255) / 256, 256, 0, s>>>(MAXA, 0x007FFFFFu, Nd * 2);  // flip(-inf)
    fill_f32<<<(Nd * 2 + 255) / 256, 256, 0, s>>>(DEN, 0.0f, Nd * 2);
    fill_f32<<<(Nd * DD + 255) / 256, 256, 0, s>>>(AGG, 0.0f, Nd * DD);
    edge_alpha<<<(E + 7) / 8, 256, 0, s>>>(ei, E, Kb, Qb, p.p_rel, ALPHA, MAXA);
    edge_expsum<<<(E * 2 + 255) / 256, 256, 0, s>>>(ei, E, ALPHA, MAXA, DEN);
    edge_agg<<<(E * DD + 255) / 256, 256, 0, s>>>(ei, E, ALPHA, DEN, Vb, AGG);
    // out = gate*(gelu(agg) @ W_out + b_out) + (1-gate)*x_dst  [+extra][+=]
    gelu_f32<<<(Nd * DD + 255) / 256, 256, 0, s>>>(Qb, AGG, Nd * DD);  // Q no longer needed
    launch_gemm(s, Qb, p.W_out, DD, p.b_out, Yregion, DD, p.skip, xdst, extra, accum, Nd,
                DD, DD, DD);
}

extern "C" void kernel_launch(void* const* d_in, const int* in_sizes, int n_in,
                              void* d_out, int out_size, void* d_ws, size_t ws_size,
                              hipStream_t stream) {
    (void)in_sizes; (void)n_in; (void)out_size; (void)ws_size;
    auto F = [&](int i) { return (const float*)d_in[i]; };
    auto I = [&](int i) { return (const int*)d_in[i]; };

    const float* x_leg  = F(0);
    const float* x_bill = F(1);
    const float* x_bv   = F(2);
    const float* x_don  = F(3);
    const float* x_lob  = F(4);
    const int *ei_don = I(5), *ei_lob = I(6), *ei_ver = I(7), *ei_vote = I(8);
    const float* attr   = F(9);
    const float *ts_don = F(10), *ts_lob = F(11), *ts_vote = F(12);

    auto conv = [&](int b) {
        ConvP p;
        p.W_src = F(b); p.b_src = F(b + 1); p.W_dst = F(b + 2); p.b_dst = F(b + 3);
        p.a_rel = F(b + 4); p.m_rel = F(b + 5); p.p_rel = F(b + 6);
        p.W_out = F(b + 7); p.b_out = F(b + 8); p.skip = F(b + 9);
        return p;
    };
    ConvP pdon = conv(13), plob = conv(23), pver = conv(33), pvote = conv(43);
    const float *tdW1 = F(53), *tdb1 = F(54), *tdW2 = F(55), *tdb2 = F(56);
    const float *tlW1 = F(57), *tlb1 = F(58), *tlW2 = F(59), *tlb2 = F(60);
    const float *tvW1 = F(61), *tvb1 = F(62), *tvW2 = F(63), *tvb2 = F(64);
    const float *vmW1 = F(65), *vmb1 = F(66), *vmW2 = F(67), *vmb2 = F(68);

    float* out = (float*)d_out;
    float* o_leg  = out;
    float* o_bill = out + (size_t)NLEG * DD;
    float* o_bv   = o_bill + (size_t)NBILL * DD;
    float* o_don  = o_bv + (size_t)NBV * DD;
    float* o_lob  = o_don + (size_t)NDON * DD;

    // workspace layout (floats)
    float*    ws      = (float*)d_ws;
    float*    TA_DON  = ws;                                  // 50000*192
    float*    TA_LOB  = TA_DON + (size_t)NDON * DD;          // 20000*192
    float*    TA_VOTE = TA_LOB + (size_t)NLOB * DD;          // 20000*192
    float*    Kb      = TA_VOTE + (size_t)NLEG * DD;         // 100000*192
    float*    Vb      = Kb + (size_t)NBV * DD;
    float*    Qb      = Vb + (size_t)NBV * DD;
    float*    AGG     = Qb + (size_t)NBV * DD;
    unsigned* MAXA    = (unsigned*)(AGG + (size_t)NBV * DD); // 100000*2
    float*    DEN     = (float*)(MAXA + (size_t)NBV * 2);    // 100000*2
    float*    ALPHA   = DEN + (size_t)NBV * 2;               // 300000*2
    float*    CNT     = ALPHA + (size_t)EDON * 2;            // 100000

    // pass-through nodes
    (void)hipMemcpyAsync(o_don, x_don, sizeof(float) * (size_t)NDON * DD,
                         hipMemcpyDeviceToDevice, stream);
    (void)hipMemcpyAsync(o_lob, x_lob, sizeof(float) * (size_t)NLOB * DD,
                         hipMemcpyDeviceToDevice, stream);

    // ---- temporal adjustments
    int taTot = (NDON + NLOB + NLEG) * DD;
    fill_f32<<<(taTot + 255) / 256, 256, 0, stream>>>(TA_DON, 0.0f, taTot);
    fill_f32<<<(NDON + 255) / 256, 256, 0, stream>>>(CNT, 0.0f, NDON);
    temporal_edge<<<EDON, DD, 0, stream>>>(ts_don, ei_don, EDON, tdW1, tdb1, tdW2, tdb2,
                                           TA_DON, CNT);
    ta_finalize<<<(NDON * DD + 255) / 256, 256, 0, stream>>>(TA_DON, CNT, NDON);

    fill_f32<<<(NLOB + 255) / 256, 256, 0, stream>>>(CNT, 0.0f, NLOB);
    temporal_edge<<<ELOB, DD, 0, stream>>>(ts_lob, ei_lob, ELOB, tlW1, tlb1, tlW2, tlb2,
                                           TA_LOB, CNT);
    ta_finalize<<<(NLOB * DD + 255) / 256, 256, 0, stream>>>(TA_LOB, CNT, NLOB);

    fill_f32<<<(NLEG + 255) / 256, 256, 0, stream>>>(CNT, 0.0f, NLEG);
    temporal_edge<<<EVOTE, DD, 0, stream>>>(ts_vote, ei_vote, EVOTE, tvW1, tvb1, tvW2,
                                            tvb2, TA_VOTE, CNT);
    ta_finalize<<<(NLEG * DD + 255) / 256, 256, 0, stream>>>(TA_VOTE, CNT, NLEG);

    // ---- four HGT convolutions
    // out_leg = hgt_don + hgt_lob + x_leg
    run_hgt(stream, pdon, x_don, TA_DON, NDON, x_leg, NLEG, ei_don, EDON, o_leg, x_leg, 0,
            Kb, Vb, Qb, AGG, MAXA, DEN, ALPHA);
    run_hgt(stream, plob, x_lob, TA_LOB, NLOB, x_leg, NLEG, ei_lob, ELOB, o_leg, 0, 1, Kb,
            Vb, Qb, AGG, MAXA, DEN, ALPHA);
    // out_bill = hgt_ver + x_bill
    run_hgt(stream, pver, x_bv, 0, NBV, x_bill, NBILL, ei_ver, EVER, o_bill, x_bill, 0,
            Kb, Vb, Qb, AGG, MAXA, DEN, ALPHA);
    // out_bv = hgt_vote + x_bv (+ polarity conv below)
    run_hgt(stream, pvote, x_leg, TA_VOTE, NLEG, x_bv, NBV, ei_vote, EVOTE, o_bv, x_bv, 0,
            Kb, Vb, Qb, AGG, MAXA, DEN, ALPHA);

    // ---- polarity-aware conv on vote edges, scatter-mean into bv
    fill_f32<<<(NBV * DD + 255) / 256, 256, 0, stream>>>(AGG, 0.0f, NBV * DD);
    fill_f32<<<(NBV + 255) / 256, 256, 0, stream>>>(CNT, 0.0f, NBV);
    polarity_conv<<<EVOTE / 16, 128, 0, stream>>>(attr, ei_vote, EVOTE, vmW1, vmb1, vmW2,
                                                  vmb2, AGG, CNT);
    polarity_fin<<<(NBV * DD + 255) / 256, 256, 0, stream>>>(o_bv, AGG, CNT, NBV);
}